// HyperbolicLocalAgg_56341380989450
// MI455X (gfx1250) — compile-verified
//
#include <hip/hip_runtime.h>
#include <hip/hip_bf16.h>
#include <math.h>

#define NPTS 1024
#define DIM  64
#define EPSF 1e-7f

typedef __attribute__((ext_vector_type(2))) float v2f;
typedef __attribute__((ext_vector_type(8))) float v8f;

// ---------------------------------------------------------------------------
// Kernel 1: Gram matrix C = X * X^T  (1024x1024, K=64) via v_wmma_f32_16x16x4_f32.
// One wave per 16x16 output tile; 16 WMMA steps of K=4.
// A layout (16x4 f32): lane<16 -> M=lane, K={k0,k0+1}; lane>=16 -> K={k0+2,k0+3}.
// B layout mirrors A with N=lane%16; since B = X^T, B[k][n] = X[n][k].
// ---------------------------------------------------------------------------
__global__ __launch_bounds__(32) void gram_kernel(const float* __restrict__ X,
                                                  float* __restrict__ C) {
  const int m    = blockIdx.y;          // row tile
  const int n    = blockIdx.x;          // col tile
  const int lane = threadIdx.x;
  const int half = lane >> 4;
  const int l16  = lane & 15;
  const int arow = m * 16 + l16;        // A-matrix row (M)
  const int brow = n * 16 + l16;        // B-matrix col (N) -> row of X
  v8f acc = {};
#pragma unroll
  for (int k0 = 0; k0 < DIM; k0 += 4) {
    const int ka = k0 + half * 2;
    v2f a = *(const v2f*)(X + arow * DIM + ka);
    v2f b = *(const v2f*)(X + brow * DIM + ka);
    acc = __builtin_amdgcn_wmma_f32_16x16x4_f32(false, a, false, b,
                                                (short)0, acc, false, false);
  }
  // C/D layout: VGPR r -> M = r (lanes 0-15) or M = r+8 (lanes 16-31); N = lane%16.
  const int ncol = n * 16 + l16;
#pragma unroll
  for (int r = 0; r < 8; ++r) {
    const int mrow = m * 16 + half * 8 + r;
    C[mrow * NPTS + ncol] = acc[r];
  }
}

// ---------------------------------------------------------------------------
// Kernel 2: norms[i] = ||x_i||^2 = C[i][i]
// ---------------------------------------------------------------------------
__global__ void norms_kernel(const float* __restrict__ C, float* __restrict__ norms) {
  const int i = blockIdx.x * blockDim.x + threadIdx.x;
  if (i < NPTS) norms[i] = C[i * NPTS + i];
}

// ---------------------------------------------------------------------------
// Kernel 3: per-pair scalar coefficients. In-place C -> Q (each element read
// then written by the same thread). Row sums of p are folded into Q[i][i].
//   log_{x_i}(x_j) = p_ij x_i + q_ij x_j
// ---------------------------------------------------------------------------
__global__ __launch_bounds__(256) void coef_kernel(const float* Cin,
                                                   const int* __restrict__ adj,
                                                   const float* __restrict__ norms,
                                                   float* Q) {
  __shared__ float red[256];
  const int i   = blockIdx.x;
  const int tid = threadIdx.x;
  const float ni   = norms[i];
  const float omni = 1.0f - ni;                 // 1 - ||x_i||^2 (mobius numerator)
  const float g0   = fmaxf(omni, EPSF);         // clip(1-ni, EPS) = 2/lambda_x
  float psum = 0.0f;
  for (int j = tid; j < NPTS; j += 256) {
    const float c  = Cin[i * NPTS + j];
    const float nj = norms[j];
    const int   a  = adj[i * NPTS + j];
    const float den   = fmaxf(1.0f - 2.0f * c + ni * nj, EPSF);
    const float alpha = -(1.0f - 2.0f * c + nj) / den;
    const float beta  = omni / den;
    // ||u||^2 where u = alpha*x_i + beta*x_j
    float un2 = alpha * alpha * ni + 2.0f * alpha * beta * c + beta * beta * nj;
    float un  = sqrtf(fmaxf(un2, 0.0f));
    un = fminf(fmaxf(un, EPSF), 1.0f - 1e-5f);  // clip like reference
    // atanh(un) = 0.5*log((1+un)/(1-un))
    const float g = g0 * 0.5f * logf((1.0f + un) / (1.0f - un)) / un;
    float p = g * alpha;
    float q = g * beta;
    if (a == 0) { p = 0.0f; q = 0.0f; }
    Q[i * NPTS + j] = q;
    psum += p;
  }
  red[tid] = psum;
  __syncthreads();
  for (int s = 128; s > 0; s >>= 1) {
    if (tid < s) red[tid] += red[tid + s];
    __syncthreads();
  }
  if (tid == 0) Q[i * NPTS + i] += red[0];      // fold p-rowsum: + p_i * x_i
}

// ---------------------------------------------------------------------------
// Kernel 4: V = Q' @ X  (1024x64, K=1024) via v_wmma_f32_16x16x4_f32.
// One wave per 16x16 tile of V; 256 WMMA steps of K=4.
// B[k][n] = X[k][col]; loads across lanes of one K-row are coalesced.
// ---------------------------------------------------------------------------
__global__ __launch_bounds__(32) void agg_kernel(const float* __restrict__ Q,
                                                 const float* __restrict__ X,
                                                 float* __restrict__ V) {
  const int m    = blockIdx.y;          // 0..63 row tile
  const int n    = blockIdx.x;          // 0..3  col tile
  const int lane = threadIdx.x;
  const int half = lane >> 4;
  const int l16  = lane & 15;
  const int arow = m * 16 + l16;
  const int col  = n * 16 + l16;
  v8f acc = {};
  for (int k0 = 0; k0 < NPTS; k0 += 4) {
    const int ka = k0 + half * 2;
    v2f a = *(const v2f*)(Q + arow * NPTS + ka);
    v2f b;
    b.x = X[ka * DIM + col];
    b.y = X[(ka + 1) * DIM + col];
    acc = __builtin_amdgcn_wmma_f32_16x16x4_f32(false, a, false, b,
                                                (short)0, acc, false, false);
  }
#pragma unroll
  for (int r = 0; r < 8; ++r) {
    const int mrow = m * 16 + half * 8 + r;
    V[mrow * DIM + col] = acc[r];
  }
}

// ---------------------------------------------------------------------------
// Kernel 5: out_i = exp_{x_i}(v_i). One wave32 per row, 2 dims per lane,
// shuffle reductions for ||v||^2 and <x,v>.
// ---------------------------------------------------------------------------
__global__ __launch_bounds__(256) void expmap_kernel(const float* __restrict__ X,
                                                     const float* __restrict__ V,
                                                     const float* __restrict__ norms,
                                                     float* __restrict__ out) {
  const int wave = threadIdx.x >> 5;
  const int lane = threadIdx.x & 31;
  const int i    = blockIdx.x * 8 + wave;
  const float2 v  = *(const float2*)(V + i * DIM + lane * 2);
  const float2 xi = *(const float2*)(X + i * DIM + lane * 2);
  float vv = v.x * v.x + v.y * v.y;       // -> ||v||^2
  float xv = xi.x * v.x + xi.y * v.y;     // -> <x, v>
#pragma unroll
  for (int off = 16; off > 0; off >>= 1) {
    vv += __shfl_xor(vv, off, 32);
    xv += __shfl_xor(xv, off, 32);
  }
  const float ni = norms[i];
  const float vn = fmaxf(sqrtf(vv), EPSF);                 // clip(||v||, EPS)
  const float t  = tanhf(vn / fmaxf(1.0f - ni, EPSF));     // tanh(lambda*vn/2)
  const float tv = t / vn;                                 // second = tv * v
  const float s2 = tv * tv * vv;                           // ||second||^2
  const float xs = tv * xv;                                // <x, second>
  const float cx  = 1.0f + 2.0f * xs + s2;
  const float cs  = 1.0f - ni;
  const float inv = 1.0f / fmaxf(1.0f + 2.0f * xs + ni * s2, EPSF);
  float2 o;
  o.x = (cx * xi.x + cs * (tv * v.x)) * inv;
  o.y = (cx * xi.y + cs * (tv * v.y)) * inv;
  *(float2*)(out + i * DIM + lane * 2) = o;
}

// ---------------------------------------------------------------------------
extern "C" void kernel_launch(void* const* d_in, const int* in_sizes, int n_in,
                              void* d_out, int out_size, void* d_ws, size_t ws_size,
                              hipStream_t stream) {
  const float* X   = (const float*)d_in[0];   // [1024, 64] f32
  const int*   adj = (const int*)d_in[1];     // [1024, 1024] i32
  float* out = (float*)d_out;                 // [1024, 64] f32

  char* ws = (char*)d_ws;
  float* C     = (float*)ws;                                   // 4 MB, reused as Q
  float* norms = (float*)(ws + (size_t)NPTS * NPTS * 4);       // 4 KB
  float* V     = (float*)(ws + (size_t)NPTS * NPTS * 4 + 4096);// 256 KB

  gram_kernel <<<dim3(NPTS / 16, NPTS / 16), 32,  0, stream>>>(X, C);
  norms_kernel<<<NPTS / 256,                 256, 0, stream>>>(C, norms);
  coef_kernel <<<NPTS,                       256, 0, stream>>>(C, adj, norms, C);
  agg_kernel  <<<dim3(DIM / 16, NPTS / 16),  32,  0, stream>>>(C, X, V);
  expmap_kernel<<<NPTS / 8,                  256, 0, stream>>>(X, V, norms, out);
}